// Interval_Softmax_56255481643386
// MI455X (gfx1250) — compile-verified
//
#include <hip/hip_runtime.h>
#include <hip/hip_bf16.h>

typedef __attribute__((ext_vector_type(4))) float v4f;

#define N_COLS 2048
#define THREADS 256
#define VPT (N_COLS / THREADS)   // 8 elements per thread per tensor
#define NWAVES (THREADS / 32)    // 8 wave32 per block

__global__ __launch_bounds__(THREADS)
void interval_softmax_kernel(const float* __restrict__ l,
                             const float* __restrict__ u,
                             float* __restrict__ out,
                             int n_rows)
{
    __shared__ float s_l[N_COLS];
    __shared__ float s_u[N_COLS];
    __shared__ float redM[NWAVES];
    __shared__ float redL[NWAVES];
    __shared__ float redU[NWAVES];

    const int tid  = threadIdx.x;
    const int row  = blockIdx.x;
    const int lane = tid & 31;
    const int wid  = tid >> 5;

    const float* lrow = l + (size_t)row * N_COLS;
    const float* urow = u + (size_t)row * N_COLS;

    // Byte offsets: thread t owns bytes [32t, 32t+32) of the row (two b128s).
    const unsigned go    = (unsigned)(tid * (VPT * 4));
    const unsigned lds_l = (unsigned)(uintptr_t)&s_l[0] + go; // low 32 bits of a
    const unsigned lds_u = (unsigned)(uintptr_t)&s_u[0] + go; // generic LDS ptr = LDS addr

    // ---- CDNA5 async copy engine: global -> LDS, tracked on ASYNCcnt ----
    // GVS mode: addr = SADDR(64b sgpr pair) + VADDR(32b voffset) + inst_offset,
    // inst_offset is added to BOTH the global and the LDS address.
    asm volatile("global_load_async_to_lds_b128 %0, %1, %2"
                 :: "v"(lds_l), "v"(go), "s"(lrow) : "memory");
    asm volatile("global_load_async_to_lds_b128 %0, %1, %2 offset:16"
                 :: "v"(lds_l), "v"(go), "s"(lrow) : "memory");
    asm volatile("global_load_async_to_lds_b128 %0, %1, %2"
                 :: "v"(lds_u), "v"(go), "s"(urow) : "memory");
    asm volatile("global_load_async_to_lds_b128 %0, %1, %2 offset:16"
                 :: "v"(lds_u), "v"(go), "s"(urow) : "memory");
    asm volatile("s_wait_asynccnt 0x0" ::: "memory");
    __syncthreads();

    // Pull my 16 values back out of LDS (b128 ds loads).
    const int base = tid * VPT;
    v4f vl0 = *(const v4f*)&s_l[base];
    v4f vl1 = *(const v4f*)&s_l[base + 4];
    v4f vu0 = *(const v4f*)&s_u[base];
    v4f vu1 = *(const v4f*)&s_u[base + 4];

    float lv[VPT] = {vl0.x, vl0.y, vl0.z, vl0.w, vl1.x, vl1.y, vl1.z, vl1.w};
    float uv[VPT] = {vu0.x, vu0.y, vu0.z, vu0.w, vu1.x, vu1.y, vu1.z, vu1.w};

    // ---- rowwise max: registers -> wave32 shuffle tree -> LDS scratch ----
    float m = lv[0];
#pragma unroll
    for (int i = 1; i < VPT; ++i) m = fmaxf(m, lv[i]);
#pragma unroll
    for (int i = 0; i < VPT; ++i) m = fmaxf(m, uv[i]);
#pragma unroll
    for (int off = 16; off > 0; off >>= 1)
        m = fmaxf(m, __shfl_xor(m, off, 32));
    if (lane == 0) redM[wid] = m;
    __syncthreads();
    float M = redM[0];
#pragma unroll
    for (int i = 1; i < NWAVES; ++i) M = fmaxf(M, redM[i]);

    // ---- exponentials (hardware v_exp_f32) + rowwise sums ----
    float el[VPT], eu[VPT];
    float sl = 0.0f, su = 0.0f;
#pragma unroll
    for (int i = 0; i < VPT; ++i) {
        el[i] = __expf(lv[i] - M);
        eu[i] = __expf(uv[i] - M);
        sl += el[i];
        su += eu[i];
    }
#pragma unroll
    for (int off = 16; off > 0; off >>= 1) {
        sl += __shfl_xor(sl, off, 32);
        su += __shfl_xor(su, off, 32);
    }
    if (lane == 0) { redL[wid] = sl; redU[wid] = su; }
    __syncthreads();
    float Sl = 0.0f, Su = 0.0f;
#pragma unroll
    for (int i = 0; i < NWAVES; ++i) { Sl += redL[i]; Su += redU[i]; }

    // ---- diag-of-softmax results; denominators are strictly positive ----
    float rl[VPT], ru[VPT];
#pragma unroll
    for (int i = 0; i < VPT; ++i) {
        rl[i] = __fdividef(el[i], el[i] + (Su - eu[i]));
        ru[i] = __fdividef(eu[i], eu[i] + (Sl - el[i]));
    }

    // ---- streaming (non-temporal) b128 stores; outputs never re-read ----
    float* outL = out + (size_t)row * N_COLS + base;
    float* outU = out + (size_t)n_rows * N_COLS + (size_t)row * N_COLS + base;
    v4f oL0 = {rl[0], rl[1], rl[2], rl[3]};
    v4f oL1 = {rl[4], rl[5], rl[6], rl[7]};
    v4f oU0 = {ru[0], ru[1], ru[2], ru[3]};
    v4f oU1 = {ru[4], ru[5], ru[6], ru[7]};
    __builtin_nontemporal_store(oL0, (v4f*)outL);
    __builtin_nontemporal_store(oL1, (v4f*)(outL + 4));
    __builtin_nontemporal_store(oU0, (v4f*)outU);
    __builtin_nontemporal_store(oU1, (v4f*)(outU + 4));
}

extern "C" void kernel_launch(void* const* d_in, const int* in_sizes, int n_in,
                              void* d_out, int out_size, void* d_ws, size_t ws_size,
                              hipStream_t stream) {
    (void)n_in; (void)out_size; (void)d_ws; (void)ws_size;
    const float* l = (const float*)d_in[0];
    const float* u = (const float*)d_in[1];
    float* out = (float*)d_out;
    const int n_rows = in_sizes[0] / N_COLS;  // 4096 for the reference shapes
    interval_softmax_kernel<<<n_rows, THREADS, 0, stream>>>(l, u, out, n_rows);
}